// SurfaceConv_76622216561208
// MI455X (gfx1250) — compile-verified
//
#include <hip/hip_runtime.h>
#include <hip/hip_bf16.h>
#include <math.h>

// ---------------- problem constants ----------------
#define BB   8
#define NN   8192
#define CC   64
#define FE   32
#define KNN  16

// ---------------- workspace layout (float slots) ----------------
#define OFF_SCALE 0            // 64
#define OFF_SHIFT 64           // 64
#define OFF_WGP   128          // 32*68 = 2176 (folded Wg', K padded to 68)
#define OFF_BIAS  2304         // 32   (contiguous after WGP on purpose)
#define OFF_WHS   2336         // 64*32 = 2048 (Wh summed over the 4 identical blocks)
#define OFF_FEAT0 4384         // 8*8192*32 = 2097152
#define OFF_IDX   2101536      // int32 region: 8*8192*16 = 1048576

typedef __attribute__((ext_vector_type(2))) float v2f;
typedef __attribute__((ext_vector_type(8))) float v8f;

// ================= Kernel A1: BN statistics -> scale/shift =================
__global__ __launch_bounds__(256) void k_bn_stats(const float* __restrict__ feat,
                                                  const float* __restrict__ gamma,
                                                  const float* __restrict__ beta,
                                                  float* __restrict__ ws) {
    const int c = blockIdx.x;      // one block per channel
    const int t = threadIdx.x;
    float s = 0.f, sq = 0.f;
    for (int i = t; i < BB * NN; i += 256) {
        int b = i >> 13;           // N == 8192
        int n = i & (NN - 1);
        float v = feat[((size_t)b * CC + c) * NN + n];
        s += v;
        sq = fmaf(v, v, sq);
    }
    __shared__ float ls[256], lq[256];
    ls[t] = s; lq[t] = sq;
    __syncthreads();
    for (int o = 128; o > 0; o >>= 1) {
        if (t < o) { ls[t] += ls[t + o]; lq[t] += lq[t + o]; }
        __syncthreads();
    }
    if (t == 0) {
        const float inv = 1.f / (float)(BB * NN);
        float mean = ls[0] * inv;
        float var  = lq[0] * inv - mean * mean;      // population var (ddof=0)
        float sc   = gamma[c] * rsqrtf(var + 1e-5f);
        ws[OFF_SCALE + c] = sc;
        ws[OFF_SHIFT + c] = beta[c] - mean * sc;
    }
}

// ============ Kernel A2: fold BN into Wg -> Wg' + bias; pre-sum Wh ============
__global__ __launch_bounds__(256) void k_prep(const float* __restrict__ Wg,
                                              const float* __restrict__ Wh,
                                              float* __restrict__ ws) {
    const int t = threadIdx.x;
    const float* scale = ws + OFF_SCALE;
    const float* shift = ws + OFF_SHIFT;
    // Wg' [32][68]: columns <64 scaled by BN scale, 64..66 pass-through (xyz), 67 zero pad
    for (int i = t; i < 32 * 68; i += 256) {
        int e = i / 68, c = i % 68;
        float v = 0.f;
        if (c < 64)      v = Wg[e * 67 + c] * scale[c];
        else if (c < 67) v = Wg[e * 67 + c];
        ws[OFF_WGP + i] = v;
    }
    // bias[e] = sum_c Wg[e,c] * shift[c]
    if (t < 32) {
        float s = 0.f;
        for (int c = 0; c < 64; ++c) s = fmaf(Wg[t * 67 + c], shift[c], s);
        ws[OFF_BIAS + t] = s;
    }
    // Whs[c][j] = sum over the 4 identical repeated blocks
    for (int i = t; i < 64 * 32; i += 256) {
        int c = i >> 5, j = i & 31;
        const float* w = Wh + c * 128 + j;
        ws[OFF_WHS + i] = w[0] + w[32] + w[64] + w[96];
    }
}

// ================= Kernel B: feat0 = cat x Wg'^T via WMMA f32 16x16x4 =================
// 8 waves/block, each wave owns one 16-row (n) tile, both 16-col (e) tiles.
__global__ __launch_bounds__(256) void k_gfunc_wmma(const float* __restrict__ feat,
                                                    const float* __restrict__ xyz,
                                                    float* __restrict__ ws) {
    __shared__ float sWg[32 * 68 + 32];            // Wg' then bias (contiguous in ws)
    const int t = threadIdx.x;
    for (int i = t; i < 32 * 68 + 32; i += 256) sWg[i] = ws[OFF_WGP + i];
    __syncthreads();

    const int wave = t >> 5, lane = t & 31;
    const int tile = blockIdx.x * 8 + wave;        // 0..4095 = B * N/16
    const int b  = tile >> 9;                      // 512 n-tiles per batch
    const int n0 = (tile & 511) << 4;
    const int l15 = lane & 15;                     // A: M row, B/D: N col
    const int kb  = (lane >> 4) << 1;              // K sub-pair: 0 or 2
    const int n   = n0 + l15;

    const float* fb = feat + (size_t)b * CC * NN;
    const float* xb = xyz  + (size_t)b * NN * 3;

    v8f acc0 = {};                                 // e-tile 0 (cols 0..15)
    v8f acc1 = {};                                 // e-tile 1 (cols 16..31)
    for (int kk = 0; kk < 17; ++kk) {              // K = 68, 4 per step
        const int ca = kk * 4 + kb;
        const int cb = ca + 1;
        // raw concat(feat, xyz^T, 0-pad) — BN already folded into Wg'
        float a0 = (ca < 64) ? fb[ca * NN + n] : ((ca < 67) ? xb[n * 3 + (ca - 64)] : 0.f);
        float a1 = (cb < 64) ? fb[cb * NN + n] : ((cb < 67) ? xb[n * 3 + (cb - 64)] : 0.f);
        v2f A  = {a0, a1};
        v2f B0 = {sWg[l15 * 68 + ca],        sWg[l15 * 68 + cb]};
        v2f B1 = {sWg[(l15 + 16) * 68 + ca], sWg[(l15 + 16) * 68 + cb]};
        acc0 = __builtin_amdgcn_wmma_f32_16x16x4_f32(false, A, false, B0, (short)0, acc0, false, false);
        acc1 = __builtin_amdgcn_wmma_f32_16x16x4_f32(false, A, false, B1, (short)0, acc1, false, false);
    }

    const float bias0 = sWg[32 * 68 + l15];
    const float bias1 = sWg[32 * 68 + 16 + l15];
    const int rowbase = (lane >> 4) << 3;          // D rows: 0..7 or 8..15
    float* f0 = ws + OFF_FEAT0 + (size_t)b * NN * FE;
#pragma unroll
    for (int v = 0; v < 8; ++v) {
        const int nn = n0 + rowbase + v;
        f0[nn * FE + l15]      = acc0[v] + bias0;
        f0[nn * FE + 16 + l15] = acc1[v] + bias1;
    }
}

// ================= Kernel C: exact KNN (top-16 smallest squared dist) =================
// One query per thread; candidates staged through LDS; register insertion list.
__global__ __launch_bounds__(256) void k_knn(const float* __restrict__ xyz,
                                             float* __restrict__ ws) {
    __shared__ float sx[256], sy[256], sz[256];
    const int t = threadIdx.x;
    const int b = blockIdx.x >> 5;                 // 32 blocks per batch
    const int n = ((blockIdx.x & 31) << 8) + t;
    const float* xb = xyz + (size_t)b * NN * 3;
    const float qx = xb[n * 3 + 0];
    const float qy = xb[n * 3 + 1];
    const float qz = xb[n * 3 + 2];

    float bd[KNN];
    int   bi[KNN];
#pragma unroll
    for (int i = 0; i < KNN; ++i) { bd[i] = 3.4e38f; bi[i] = 0; }

    for (int j0 = 0; j0 < NN; j0 += 256) {
        __syncthreads();
        sx[t] = xb[(j0 + t) * 3 + 0];
        sy[t] = xb[(j0 + t) * 3 + 1];
        sz[t] = xb[(j0 + t) * 3 + 2];
        __syncthreads();
        for (int j = 0; j < 256; ++j) {
            float dx = qx - sx[j];
            float dy = qy - sy[j];
            float dz = qz - sz[j];
            float d2 = fmaf(dx, dx, fmaf(dy, dy, dz * dz));
            if (d2 < bd[KNN - 1]) {                // strict < keeps earliest index on ties
                bd[KNN - 1] = d2; bi[KNN - 1] = j0 + j;
#pragma unroll
                for (int s = KNN - 1; s > 0; --s) {
                    if (bd[s] < bd[s - 1]) {
                        float td = bd[s]; bd[s] = bd[s - 1]; bd[s - 1] = td;
                        int   ti = bi[s]; bi[s] = bi[s - 1]; bi[s - 1] = ti;
                    } else break;
                }
            }
        }
    }
    int* idxw = (int*)(ws + OFF_IDX);
#pragma unroll
    for (int i = 0; i < KNN; ++i)
        idxw[((size_t)b * NN + n) * KNN + i] = bi[i];
}

// ====== Kernel D: gather + max-pool + rel, then out = rel x Whs^T via WMMA ======
// Block = one 16-query tile; waves 0..3 each produce one 16-channel tile.
__global__ __launch_bounds__(256) void k_hfunc_wmma(float* __restrict__ ws,
                                                    float* __restrict__ out) {
    __shared__ float sWhs[64 * 32];
    __shared__ float sRel[16 * 32];
    __shared__ int   sIdx[256];

    const int t = threadIdx.x;
    const int tile = blockIdx.x;                   // 0..4095
    const int b  = tile >> 9;
    const int n0 = (tile & 511) << 4;

    for (int i = t; i < 64 * 32; i += 256) sWhs[i] = ws[OFF_WHS + i];
    const int* idxw = (const int*)(ws + OFF_IDX);
    sIdx[t] = idxw[((size_t)b * NN + n0) * KNN + t];   // 16 queries x 16 neighbors
    __syncthreads();

    // rel: thread (q, fp) handles features 2*fp, 2*fp+1 of query q
    const float* f0 = ws + OFF_FEAT0 + (size_t)b * NN * FE;
    {
        const int q = t >> 4, fp = t & 15;
        float g0 = -3.4e38f, g1 = -3.4e38f;
#pragma unroll
        for (int k = 0; k < KNN; ++k) {
            const int j = sIdx[q * KNN + k];
            g0 = fmaxf(g0, f0[j * FE + 2 * fp]);
            g1 = fmaxf(g1, f0[j * FE + 2 * fp + 1]);
        }
        sRel[q * FE + 2 * fp]     = g0 - f0[(n0 + q) * FE + 2 * fp];
        sRel[q * FE + 2 * fp + 1] = g1 - f0[(n0 + q) * FE + 2 * fp + 1];
    }
    __syncthreads();

    const int wave = t >> 5, lane = t & 31;
    if (wave < 4) {                                 // wave-uniform: EXEC all-1 inside
        const int c0  = wave << 4;
        const int l15 = lane & 15;
        const int kb  = (lane >> 4) << 1;
        v8f acc = {};
#pragma unroll
        for (int kk = 0; kk < 8; ++kk) {            // K = 32
            const int ka = kk * 4 + kb;
            v2f A  = {sRel[l15 * FE + ka],         sRel[l15 * FE + ka + 1]};
            v2f Bm = {sWhs[(c0 + l15) * FE + ka],  sWhs[(c0 + l15) * FE + ka + 1]};
            acc = __builtin_amdgcn_wmma_f32_16x16x4_f32(false, A, false, Bm, (short)0, acc, false, false);
        }
        const int rowbase = (lane >> 4) << 3;       // n sub-rows 0..7 / 8..15
        float* ob = out + ((size_t)b * CC + (c0 + l15)) * NN + n0 + rowbase;
        *(float4*)(ob)     = make_float4(acc[0], acc[1], acc[2], acc[3]);
        *(float4*)(ob + 4) = make_float4(acc[4], acc[5], acc[6], acc[7]);
    }
}

// ================================ launcher ================================
extern "C" void kernel_launch(void* const* d_in, const int* in_sizes, int n_in,
                              void* d_out, int out_size, void* d_ws, size_t ws_size,
                              hipStream_t stream) {
    const float* xyz   = (const float*)d_in[0];
    const float* feat  = (const float*)d_in[1];
    const float* gamma = (const float*)d_in[2];
    const float* beta  = (const float*)d_in[3];
    const float* Wg    = (const float*)d_in[4];
    const float* Wh    = (const float*)d_in[5];
    float* out = (float*)d_out;
    float* ws  = (float*)d_ws;

    k_bn_stats  <<<CC,                  256, 0, stream>>>(feat, gamma, beta, ws);
    k_prep      <<<1,                   256, 0, stream>>>(Wg, Wh, ws);
    k_gfunc_wmma<<<(BB * NN / 16) / 8,  256, 0, stream>>>(feat, xyz, ws);
    k_knn       <<<BB * NN / 256,       256, 0, stream>>>(xyz, ws);
    k_hfunc_wmma<<<BB * NN / 16,        256, 0, stream>>>(ws, out);
}